// SSDTableBatchedEmbeddingBags_88149908783405
// MI455X (gfx1250) — compile-verified
//
#include <hip/hip_runtime.h>
#include <hip/hip_bf16.h>
#include <utility>

// Table-batched embedding bag, SUM pooling -> [B, T*D]
// HBM-bandwidth-bound random gather (~335 MB of 512B rows, ~15us roofline at
// 23.3 TB/s). CDNA5 path: async global->LDS copies (GLOBAL_LOAD_ASYNC_TO_LDS_
// B128, ASYNCcnt) keep 20 row-gathers in flight per wave; in-order completion
// lets us drain incrementally with descending s_wait_asynccnt immediates.

constexpr int kDim           = 128;       // embedding dim (floats)
constexpr int kRowBytes      = kDim * 4;  // 512 B per row
constexpr int kChunk         = 20;        // rows staged per drain (reference L)
constexpr int kWavesPerBlock = 4;
constexpr int kThreads       = kWavesPerBlock * 32;

__device__ __forceinline__ void async_row_to_lds(unsigned lds_byte_addr,
                                                 unsigned long long gaddr) {
  // Per-lane: copies 16 bytes from the global address gaddr to the LDS address lds_byte_addr. 32 lanes = 512 B row.
  asm volatile("global_load_async_to_lds_b128 %0, %1, off"
               :: "v"(lds_byte_addr), "v"(gaddr)
               : "memory");
}

template <int N>
__device__ __forceinline__ void wait_async_le() {
  asm volatile("s_wait_asynccnt %0" :: "n"(N) : "memory");
}

__device__ __forceinline__ void wait_async_zero() {
  asm volatile("s_wait_asynccnt 0x0" ::: "memory");
}

// Consume row R from LDS as soon as its (in-order) async copy has landed:
// after issuing kChunk copies, ASYNCcnt <= kChunk-1-R  =>  rows 0..R complete.
template <int R>
__device__ __forceinline__ void consume_row(const float* smrow_base, int lane,
                                            float4& acc) {
  wait_async_le<kChunk - 1 - R>();
  const float4 v =
      *(const float4*)(smrow_base + (size_t)R * kDim + (size_t)lane * 4);
  acc.x += v.x; acc.y += v.y; acc.z += v.z; acc.w += v.w;
}

template <int... Rs>
__device__ __forceinline__ void drain_all(const float* smrow_base, int lane,
                                          float4& acc,
                                          std::integer_sequence<int, Rs...>) {
  (consume_row<Rs>(smrow_base, lane, acc), ...);
}

__global__ __launch_bounds__(kThreads, 1)
void tbe_pool_async_kernel(const int* __restrict__ indices,
                           const int* __restrict__ offsets,
                           const float* __restrict__ weights,
                           const int* __restrict__ hash_cumsum,
                           float* __restrict__ out,
                           int batch, int n_tables) {
  __shared__ float smem[kWavesPerBlock][kChunk][kDim];  // 40 KB

  // blockIdx.y == table  =>  no integer division; bag math is all-scalar.
  const int wave  = __builtin_amdgcn_readfirstlane((int)(threadIdx.x >> 5));
  const int lane  = (int)(threadIdx.x & 31);
  const int table = (int)blockIdx.y;
  const int b     = (int)blockIdx.x * kWavesPerBlock + wave;  // wave-uniform
  if (b >= batch) return;                                     // uniform exit
  const int bag = table * batch + b;

  const long long base_row = (long long)hash_cumsum[table];
  const int start = offsets[bag];      // scalar loads (uniform address)
  const int end   = offsets[bag + 1];

  const unsigned long long wbase = (unsigned long long)(uintptr_t)weights;
  const unsigned lane_byte = (unsigned)lane * 16u;
  // Low 32 bits of the generic shared-pointer == LDS byte offset (aperture rule).
  const unsigned lds_base =
      (unsigned)(unsigned long long)(uintptr_t)(&smem[wave][0][0]) + lane_byte;
  const float* smrow_base = &smem[wave][0][0];

  float4 acc = make_float4(0.f, 0.f, 0.f, 0.f);

  if (end - start == kChunk) {
    // ---- Fast path: fixed bag size, fully unrolled ----
    long long rows[kChunk];
#pragma unroll
    for (int r = 0; r < kChunk; ++r)   // batched scalar index loads
      rows[r] = base_row + (long long)indices[start + r];
#pragma unroll
    for (int r = 0; r < kChunk; ++r) { // 20 async 512B gathers in flight
      const unsigned long long gaddr =
          wbase + (unsigned long long)rows[r] * kRowBytes + lane_byte;
      async_row_to_lds(lds_base + (unsigned)(r * kRowBytes), gaddr);
    }
    // Incremental drain: overlap accumulation with in-flight gathers.
    drain_all(smrow_base, lane, acc,
              std::make_integer_sequence<int, kChunk>{});
  } else {
    // ---- Generic path: stage up to kChunk rows per full drain ----
    for (int r0 = start; r0 < end; r0 += kChunk) {
      const int n = min(end - r0, kChunk);
      for (int r = 0; r < n; ++r) {
        const long long row = base_row + (long long)indices[r0 + r];
        const unsigned long long gaddr =
            wbase + (unsigned long long)row * kRowBytes + lane_byte;
        async_row_to_lds(lds_base + (unsigned)(r * kRowBytes), gaddr);
      }
      wait_async_zero();
      for (int r = 0; r < n; ++r) {
        const float4 v = *(const float4*)&smem[wave][r][lane * 4];
        acc.x += v.x; acc.y += v.y; acc.z += v.z; acc.w += v.w;
      }
    }
  }

  // out[b, table*kDim + lane*4 .. +3] — coalesced 512 B store per bag
  float* dst = out + (size_t)b * ((size_t)n_tables * kDim)
                   + (size_t)table * kDim + (size_t)lane * 4;
  *(float4*)dst = acc;
}

extern "C" void kernel_launch(void* const* d_in, const int* in_sizes, int n_in,
                              void* d_out, int out_size, void* d_ws, size_t ws_size,
                              hipStream_t stream) {
  const int*   indices     = (const int*)d_in[0];
  const int*   offsets     = (const int*)d_in[1];
  const float* weights     = (const float*)d_in[2];
  const int*   hash_cumsum = (const int*)d_in[3];
  float*       out         = (float*)d_out;

  const int num_bags = in_sizes[1] - 1;      // T*B
  const int n_tables = in_sizes[3] - 1;      // T
  const int batch    = num_bags / n_tables;  // B

  dim3 grid((batch + kWavesPerBlock - 1) / kWavesPerBlock, n_tables);
  tbe_pool_async_kernel<<<grid, kThreads, 0, stream>>>(
      indices, offsets, weights, hash_cumsum, out, batch, n_tables);
}